// MPNN_31456340476250
// MI455X (gfx1250) — compile-verified
//
#include <hip/hip_runtime.h>
#include <hip/hip_bf16.h>
#include <math.h>

// ---------------------------------------------------------------------------
// 2-layer GCN (project -> scatter-add -> relu -> project -> scatter-add ->
// log_softmax) for MI455X / gfx1250.
//
// Roofline: ~2.5 GFLOP vs ~1 GB of traffic -> memory/atomic bound at 23.3TB/s
// (~45us floor). So we keep full fp32 precision and use V_WMMA_F32_16X16X4_F32
// for the dense projections (wave32, 16x16 output tile per wave, K-loop).
// Aggregations use global f32 atomics; the destination arrays (25.6MB / 12.8MB)
// fit in the 192MB L2 so atomics resolve on-chip.
// ---------------------------------------------------------------------------

typedef __attribute__((ext_vector_type(2))) float v2f;
typedef __attribute__((ext_vector_type(8))) float v8f;

// out[M, Nc] = act(A[M, K]) @ W[Nc, K]^T + bias[Nc]
// One wave32 computes one 16x16 output tile via V_WMMA_F32_16X16X4_F32.
// A fragment (16x4, f32): lane L -> row (L&15), ks = kb + (L>>4)*2 + {0,1}
// B fragment (4x16, f32): VGPR r -> K = r + (L>>4)*2, N = L&15  (B = W^T, so
//   both fragments are contiguous float2 loads along K).
// C/D (16x16 f32, 8 VGPRs): VGPR r -> row r + (L>>4)*8, col L&15.
template <bool RELU_A>
__global__ void gcn_gemm_wmma(const float* __restrict__ A,
                              const float* __restrict__ W,
                              const float* __restrict__ bias,
                              float* __restrict__ out,
                              int M, int Nc, int K, int tilesN) {
  const int tilesM = (M + 15) >> 4;
  const int tile = blockIdx.x * blockDim.y + threadIdx.y;
  if (tile >= tilesM * tilesN) return;  // wave-uniform: EXEC stays all-1s

  const int tm = tile / tilesN;
  const int tn = tile % tilesN;

  const int lane = threadIdx.x;  // 0..31
  const int half = lane >> 4;    // 0 or 1
  const int lm = lane & 15;

  int rowA = tm * 16 + lm;
  if (rowA >= M) rowA = M - 1;  // clamp loads; stores are guarded below
  const int colN = tn * 16 + lm;  // output column == row of W

  const float* __restrict__ aPtr = A + (size_t)rowA * K + half * 2;
  const float* __restrict__ wPtr = W + (size_t)colN * K + half * 2;

  v8f acc = {};
  for (int kb = 0; kb < K; kb += 4) {
    v2f a = *(const v2f*)(aPtr + kb);
    v2f b = *(const v2f*)(wPtr + kb);
    if (RELU_A) {
      a.x = fmaxf(a.x, 0.0f);
      a.y = fmaxf(a.y, 0.0f);
    }
    // D = A(16x4) * B(4x16) + C  -- v_wmma_f32_16x16x4_f32
    acc = __builtin_amdgcn_wmma_f32_16x16x4_f32(
        /*neg_a=*/false, a, /*neg_b=*/false, b,
        /*c_mod=*/(short)0, acc, /*reuse_a=*/false, /*reuse_b=*/false);
  }

  const float bv = bias[colN];
#pragma unroll
  for (int r = 0; r < 8; ++r) {
    const int orow = tm * 16 + r + half * 8;
    if (orow < M) out[(size_t)orow * Nc + tn * 16 + lm] = acc[r] + bv;
  }
}

// agg[col[e], :] += h[row[e], :]  -- one wave per edge, lanes stride features.
__global__ void gcn_aggregate(const float* __restrict__ h,
                              const int* __restrict__ row,
                              const int* __restrict__ col,
                              float* __restrict__ agg,
                              int E, int F) {
  const int wavesPerBlock = blockDim.x >> 5;
  const int e = blockIdx.x * wavesPerBlock + (threadIdx.x >> 5);
  if (e >= E) return;
  const int lane = threadIdx.x & 31;
  const int r = row[e];
  const int c = col[e];
  const float* __restrict__ src = h + (size_t)r * F;
  float* __restrict__ dst = agg + (size_t)c * F;
  for (int f = lane; f < F; f += 32) {
    atomicAdd(dst + f, src[f]);  // global_atomic_add_f32, hits in L2
  }
}

// Row-wise log_softmax, in place. One wave per row (F <= a few * 32).
__global__ void log_softmax_rows(float* __restrict__ h, int N, int F) {
  const int wavesPerBlock = blockDim.x >> 5;
  const int rowi = blockIdx.x * wavesPerBlock + (threadIdx.x >> 5);
  if (rowi >= N) return;
  const int lane = threadIdx.x & 31;
  float* __restrict__ p = h + (size_t)rowi * F;

  float m = -INFINITY;
  for (int f = lane; f < F; f += 32) m = fmaxf(m, p[f]);
#pragma unroll
  for (int off = 16; off > 0; off >>= 1) m = fmaxf(m, __shfl_xor(m, off, 32));

  float s = 0.0f;
  for (int f = lane; f < F; f += 32) s += expf(p[f] - m);
#pragma unroll
  for (int off = 16; off > 0; off >>= 1) s += __shfl_xor(s, off, 32);

  const float lse = m + logf(s);
  for (int f = lane; f < F; f += 32) p[f] = p[f] - lse;
}

extern "C" void kernel_launch(void* const* d_in, const int* in_sizes, int n_in,
                              void* d_out, int out_size, void* d_ws,
                              size_t ws_size, hipStream_t stream) {
  const float* x = (const float*)d_in[0];
  const int* ei = (const int*)d_in[1];  // harness ABI: integer -> int32
  const float* W1 = (const float*)d_in[2];
  const float* b1 = (const float*)d_in[3];
  const float* W2 = (const float*)d_in[4];
  const float* b2 = (const float*)d_in[5];

  const int HID = in_sizes[3];            // 128
  const int OUT = in_sizes[5];            // 64
  const int IN = in_sizes[2] / HID;       // 128
  const int N = in_sizes[0] / IN;         // 50000
  const int E = in_sizes[1] / 2;          // 600000
  const int* row = ei;                    // edge_index[0]
  const int* col = ei + E;                // edge_index[1]

  // Workspace layout (fp32): h1[N,HID] | agg1[N,HID] | h2[N,OUT]  (~64 MB)
  float* h1 = (float*)d_ws;
  float* agg1 = h1 + (size_t)N * HID;
  float* h2 = agg1 + (size_t)N * HID;
  float* agg2 = (float*)d_out;  // aggregate layer 2 directly into d_out

  const dim3 gblk(32, 4);  // 4 waves / block

  // ---- Layer 1: h1 = x @ W1^T + b1 ----
  {
    const int tiles = ((N + 15) / 16) * (HID / 16);
    gcn_gemm_wmma<false><<<dim3((tiles + 3) / 4), gblk, 0, stream>>>(
        x, W1, b1, h1, N, HID, IN, HID / 16);
  }
  // Self-loop contribution: agg1 = h1, then scatter-add the E edges.
  hipMemcpyAsync(agg1, h1, (size_t)N * HID * sizeof(float),
                 hipMemcpyDeviceToDevice, stream);
  gcn_aggregate<<<dim3((E + 7) / 8), dim3(256), 0, stream>>>(h1, row, col,
                                                             agg1, E, HID);

  // ---- Layer 2: h2 = relu(agg1) @ W2^T + b2 (ReLU fused into A-load) ----
  {
    const int tiles = ((N + 15) / 16) * (OUT / 16);
    gcn_gemm_wmma<true><<<dim3((tiles + 3) / 4), gblk, 0, stream>>>(
        agg1, W2, b2, h2, N, OUT, HID, OUT / 16);
  }
  hipMemcpyAsync(agg2, h2, (size_t)N * OUT * sizeof(float),
                 hipMemcpyDeviceToDevice, stream);
  gcn_aggregate<<<dim3((E + 7) / 8), dim3(256), 0, stream>>>(h2, row, col,
                                                             agg2, E, OUT);

  // ---- log_softmax over rows of d_out, in place ----
  log_softmax_rows<<<dim3((N + 7) / 8), dim3(256), 0, stream>>>(agg2, N, OUT);
}